// VolumeRendering_13967233646656
// MI455X (gfx1250) — compile-verified
//
#include <hip/hip_runtime.h>
#include <hip/hip_bf16.h>
#include <stdint.h>

typedef float v4f __attribute__((ext_vector_type(4)));

#define N_RAYS    262144
#define N_SAMPLES 128
#define WPB       8                 // waves per block (1 ray per wave32)
#define BLOCK     (WPB * 32)

__global__ __launch_bounds__(BLOCK) void volrend_kernel(
    const float* __restrict__ sigma,
    const float* __restrict__ colors,
    const float* __restrict__ z_vals,
    const float* __restrict__ rays_d,
    float* __restrict__ out)
{
    // 8 waves * 128 samples * 3 channels * 4B = 12 KB LDS (colors staging)
    __shared__ __align__(16) float sh_colors[WPB * N_SAMPLES * 3];

    const int lane = threadIdx.x & 31;
    const int wv   = threadIdx.x >> 5;
    const int r    = blockIdx.x * WPB + wv;          // grid is exact: always valid
    const size_t rowS = (size_t)r * N_SAMPLES;

    // ---- async stage colors row (1536 B) into this wave's LDS slice --------
    // 3 issues x 32 lanes x 16B = 1536B. Tracked by ASYNCcnt.
    {
        unsigned lds_base = (unsigned)(uintptr_t)(void*)&sh_colors[wv * (N_SAMPLES * 3)];
        const char* grow  = (const char*)(colors + rowS * 3);
        #pragma unroll
        for (int k = 0; k < 3; ++k) {
            unsigned e     = (unsigned)lane + 32u * k;
            unsigned lds_a = lds_base + e * 16u;
            unsigned long long ga = (unsigned long long)(uintptr_t)(grow + e * 16u);
            asm volatile("global_load_async_to_lds_b128 %0, %1, off"
                         :: "v"(lds_a), "v"(ga) : "memory");
        }
    }

    // ---- ray direction norm (uniform per wave, hits cache) -----------------
    const float rx = rays_d[(size_t)r * 3 + 0];
    const float ry = rays_d[(size_t)r * 3 + 1];
    const float rz = rays_d[(size_t)r * 3 + 2];
    const float nrm = sqrtf(rx * rx + ry * ry + rz * rz);

    // ---- z / sigma: coalesced non-temporal b128 loads (4 samples / lane) ---
    v4f zq = __builtin_nontemporal_load((const v4f*)(z_vals + rowS) + lane);
    v4f sq = __builtin_nontemporal_load((const v4f*)(sigma  + rowS) + lane);

    // z[4i+4] comes from the next lane's zq.x (unused at lane 31: s=127 pads 1e10)
    float znext = __shfl_down(zq.x, 1, 32);

    const float d0 = (zq.y - zq.x) * nrm;
    const float d1 = (zq.z - zq.y) * nrm;
    const float d2 = (zq.w - zq.z) * nrm;
    const float d3 = ((lane == 31) ? 1e10f : (znext - zq.w)) * nrm;

    const float a0 = 1.0f - __expf(-fmaxf(sq.x, 0.0f) * d0);
    const float a1 = 1.0f - __expf(-fmaxf(sq.y, 0.0f) * d1);
    const float a2 = 1.0f - __expf(-fmaxf(sq.z, 0.0f) * d2);
    const float a3 = 1.0f - __expf(-fmaxf(sq.w, 0.0f) * d3);

    const float f0 = 1.0f - a0 + 1e-10f;
    const float f1 = 1.0f - a1 + 1e-10f;
    const float f2 = 1.0f - a2 + 1e-10f;
    const float f3 = 1.0f - a3 + 1e-10f;

    // ---- wave32 prefix product over lane-local products (ds_bpermute) ------
    float incl = ((f0 * f1) * f2) * f3;
    #pragma unroll
    for (int d = 1; d < 32; d <<= 1) {
        float t = __shfl_up(incl, d, 32);
        if (lane >= d) incl *= t;
    }
    float ex = __shfl_up(incl, 1, 32);   // exclusive scan = carry-in for this lane
    if (lane == 0) ex = 1.0f;

    // per-sample transmittance within the lane
    const float t0 = ex;
    const float t1 = t0 * f0;
    const float t2 = t1 * f1;
    const float t3 = t2 * f2;

    const float w0 = t0 * a0;
    const float w1 = t1 * a1;
    const float w2 = t2 * a2;
    const float w3 = t3 * a3;

    // ---- weights map: coalesced non-temporal b128 store --------------------
    v4f wq; wq.x = w0; wq.y = w1; wq.z = w2; wq.w = w3;
    __builtin_nontemporal_store(wq, (v4f*)(out + (size_t)3 * N_RAYS + rowS) + lane);

    // ---- colors arrive: wait ASYNCcnt, read via ds_load_b128 ---------------
    asm volatile("s_wait_asynccnt 0x0" ::: "memory");
    const v4f* shc = (const v4f*)&sh_colors[wv * (N_SAMPLES * 3) + lane * 12];
    v4f c0 = shc[0];   // s0:{x,y,z}  s1:{w,...}
    v4f c1 = shc[1];   // s1:{x,y}    s2:{z,w}
    v4f c2 = shc[2];   // s2:{x}      s3:{y,z,w}

    float racc = w0 * c0.x + w1 * c0.w + w2 * c1.z + w3 * c2.y;
    float gacc = w0 * c0.y + w1 * c1.x + w2 * c1.w + w3 * c2.z;
    float bacc = w0 * c0.z + w1 * c1.y + w2 * c2.x + w3 * c2.w;
    float acc  = (w0 + w1) + (w2 + w3);
    float dep  = (w0 * zq.x + w1 * zq.y) + (w2 * zq.z + w3 * zq.w);

    // ---- wave32 butterfly reductions (5 values) ----------------------------
    #pragma unroll
    for (int d = 16; d > 0; d >>= 1) {
        racc += __shfl_xor(racc, d, 32);
        gacc += __shfl_xor(gacc, d, 32);
        bacc += __shfl_xor(bacc, d, 32);
        acc  += __shfl_xor(acc,  d, 32);
        dep  += __shfl_xor(dep,  d, 32);
    }

    if (lane == 0) {
        // output layout: rgb[3R] | weights[128R] | depth[R] | disp[R] | acc[R]
        out[(size_t)r * 3 + 0] = racc;
        out[(size_t)r * 3 + 1] = gacc;
        out[(size_t)r * 3 + 2] = bacc;
        const size_t dbase = (size_t)N_RAYS * 131;
        out[dbase + r]                    = dep;                                  // depth
        out[dbase + N_RAYS + r]           = 1.0f / fmaxf(1e-10f, dep / acc);      // disp
        out[dbase + 2 * (size_t)N_RAYS + r] = acc;                                // acc
    }
}

extern "C" void kernel_launch(void* const* d_in, const int* in_sizes, int n_in,
                              void* d_out, int out_size, void* d_ws, size_t ws_size,
                              hipStream_t stream) {
    const float* sigma  = (const float*)d_in[0];
    const float* colors = (const float*)d_in[1];
    const float* z_vals = (const float*)d_in[2];
    const float* rays_d = (const float*)d_in[3];
    float* out = (float*)d_out;

    dim3 grid(N_RAYS / WPB);   // 32768 blocks, 8 waves (rays) each
    dim3 block(BLOCK);
    volrend_kernel<<<grid, block, 0, stream>>>(sigma, colors, z_vals, rays_d, out);
}